// VQVAE_77876347011504
// MI455X (gfx1250) — compile-verified
//
#include <hip/hip_runtime.h>
#include <hip/hip_bf16.h>

// ---------------------------------------------------------------------------
// CDNA5 / gfx1250 VQ-VAE forward.  All GEMM-like work (conv1d-as-implicit-GEMM
// and the quantizer distance matmul) runs through V_WMMA_F32_16X16X4_F32 so we
// keep full f32 precision (reference is f32) while using the matrix pipes.
// Each wave32 computes a 16(M) x 64(N) strip: 4 v8f accumulators, A-fragment
// reused across 4 WMMAs per K-step.  4 waves / 128-thread block -> 64x64 tile.
//
// Hot-loop design:
//  * reduction ordered (tap j, channel-block): tap-dependent column index +
//    pad mask hoisted out of the channel loop (no div, no EXEC divergence)
//  * software-pipelined: next iteration's A/B fragments are prefetched into a
//    second register set before the current iteration's WMMAs issue, so the
//    wave never fully drains loadcnt at the matrix op
//  * 32-bit element indices from a uniform base (SADDR-form loads, single
//    v_add_nc_u32 induction instead of 64-bit per-lane pointer adds)
//  * interior tiles (no boundary contact) take a mask-free fast path
// ---------------------------------------------------------------------------

typedef float v2f __attribute__((ext_vector_type(2)));
typedef float v8f __attribute__((ext_vector_type(8)));

#define DECAY 0.8f
#define EPSV  1e-5f

// Channel-block reduction for one tap, software-pipelined (register double
// buffering).  sc[nt] are per-lane source columns, msk[nt] the pad masks
// (used only when MASKED).
template <int KSIZE, bool MASKED>
__device__ __forceinline__ void conv_channel_loop(
    const float* __restrict__ X, const float* __restrict__ Wrow,
    int j, int Cin, int Lin, int laneHi,
    const int* sc, const float* msk, v8f* acc)
{
    const int c0 = laneHi * 2;
    unsigned woff = (unsigned)(c0 * KSIZE + j);
    unsigned xoff = (unsigned)c0 * (unsigned)Lin;
    const unsigned wstep = 4u * (unsigned)KSIZE;
    const unsigned xstep = 4u * (unsigned)Lin;

    // current fragment registers
    v2f a;
    float bx[4], by[4];
    a.x = Wrow[woff];
    a.y = Wrow[woff + KSIZE];
#pragma unroll
    for (int nt = 0; nt < 4; ++nt) {
        bx[nt] = X[xoff + (unsigned)sc[nt]];
        by[nt] = X[xoff + (unsigned)(Lin + sc[nt])];
    }

    for (int cc = 4; cc < Cin; cc += 4) {
        woff += wstep;
        xoff += xstep;
        // prefetch next iteration's fragments
        v2f an;
        float bxn[4], byn[4];
        an.x = Wrow[woff];
        an.y = Wrow[woff + KSIZE];
#pragma unroll
        for (int nt = 0; nt < 4; ++nt) {
            bxn[nt] = X[xoff + (unsigned)sc[nt]];
            byn[nt] = X[xoff + (unsigned)(Lin + sc[nt])];
        }
        // consume current fragments
#pragma unroll
        for (int nt = 0; nt < 4; ++nt) {
            v2f b;
            b.x = MASKED ? bx[nt] * msk[nt] : bx[nt];
            b.y = MASKED ? by[nt] * msk[nt] : by[nt];
            acc[nt] = __builtin_amdgcn_wmma_f32_16x16x4_f32(
                false, a, false, b, (short)0, acc[nt], false, false);
        }
        a = an;
#pragma unroll
        for (int nt = 0; nt < 4; ++nt) { bx[nt] = bxn[nt]; by[nt] = byn[nt]; }
    }
    // epilogue: last channel block
#pragma unroll
    for (int nt = 0; nt < 4; ++nt) {
        v2f b;
        b.x = MASKED ? bx[nt] * msk[nt] : bx[nt];
        b.y = MASKED ? by[nt] * msk[nt] : by[nt];
        acc[nt] = __builtin_amdgcn_wmma_f32_16x16x4_f32(
            false, a, false, b, (short)0, acc[nt], false, false);
    }
}

// ---------------------------------------------------------------------------
// Generic conv1d as implicit GEMM:
//   Y[o,t] = bias[o] + sum_{i,j} W[o,i,j] * X[i, t*STRIDE + j - pad]
// REP2 != 0 folds a nearest-neighbor 2x upsample into the input indexing
// (logical input length = 2*Lin, physical column = logical>>1).
// Fragment layouts per ISA 7.12.2 (16x4 f32 A, 4x16 B, 16x16 f32 C/D).
// ---------------------------------------------------------------------------
template <int KSIZE, int STRIDE, int REP2>
__global__ __launch_bounds__(128)
void conv1d_wmma(const float* __restrict__ X, const float* __restrict__ W,
                 const float* __restrict__ Bias, const float* __restrict__ Res,
                 float* __restrict__ Y,
                 int Cin, int Cout, int Lin, int Lout, int relu)
{
    constexpr int PAD = (KSIZE == 3) ? 1 : 0;

    const int lane    = threadIdx.x & 31;
    const int wave    = threadIdx.x >> 5;
    const int laneLow = lane & 15;
    const int laneHi  = lane >> 4;                 // 0/1: K split (A,B), M split (C,D)
    const int mBase   = (blockIdx.y * 4 + wave) * 16;
    const int nBase   = blockIdx.x * 64;
    if (mBase >= Cout) return;                     // wave-uniform exit, EXEC stays full

    const int LinLogical = REP2 ? (Lin << 1) : Lin;

    v8f acc[4] = {};

    // A fragment: lane holds row m=mBase+laneLow, channels ci0=cc+laneHi*2, +1
    const int aRow  = mBase + laneLow;
    const int aRowC = (aRow < Cout) ? aRow : (Cout - 1);
    const float* __restrict__ Wrow = W + (size_t)aRowC * (size_t)(Cin * KSIZE);

    // wave-uniform: does this 64-column tile touch the padded boundary?
    const int firstX = nBase * STRIDE - PAD;
    const int lastX  = (nBase + 63) * STRIDE + (KSIZE - 1) - PAD;
    const bool interior = (firstX >= 0) && (lastX < LinLogical);

    if (interior) {
#pragma unroll
        for (int j = 0; j < KSIZE; ++j) {
            int sc[4];
#pragma unroll
            for (int nt = 0; nt < 4; ++nt) {
                const int xcol = (nBase + nt * 16 + laneLow) * STRIDE + j - PAD;
                sc[nt] = REP2 ? (xcol >> 1) : xcol;
            }
            conv_channel_loop<KSIZE, false>(X, Wrow, j, Cin, Lin, laneHi, sc, nullptr, acc);
        }
    } else {
#pragma unroll
        for (int j = 0; j < KSIZE; ++j) {
            int   sc[4];
            float msk[4];
#pragma unroll
            for (int nt = 0; nt < 4; ++nt) {
                const int xcol = (nBase + nt * 16 + laneLow) * STRIDE + j - PAD;
                const bool ok  = (xcol >= 0) && (xcol < LinLogical);
                const int s    = REP2 ? (xcol >> 1) : xcol;
                sc[nt]  = ok ? s : 0;              // clamped safe address
                msk[nt] = ok ? 1.0f : 0.0f;        // branch-free zero padding
            }
            conv_channel_loop<KSIZE, true>(X, Wrow, j, Cin, Lin, laneHi, sc, msk, acc);
        }
    }

    // C/D layout: VGPR r, lanes 0-15 -> M=r ; lanes 16-31 -> M=r+8.
#pragma unroll
    for (int nt = 0; nt < 4; ++nt) {
        const int col = nBase + nt * 16 + laneLow;
        if (col >= Lout) continue;
#pragma unroll
        for (int r = 0; r < 8; ++r) {
            const int row = mBase + laneHi * 8 + r;
            if (row < Cout) {
                float v = acc[nt][r] + Bias[row];
                if (Res)  v += Res[(size_t)row * (size_t)Lout + col];
                if (relu) v  = v > 0.0f ? v : 0.0f;
                Y[(size_t)row * (size_t)Lout + col] = v;
            }
        }
    }
}

// ---------------------------------------------------------------------------
// Quantizer scores: S[t,k] = ||c_k||^2 - 2 * sum_d Ze[d,t]*CB[d,k]
// (||z_t||^2 is constant along k -> irrelevant for argmin; dropped.)
// A[m,kd] = Ze[kd*T + m]   (Zt tile),  B[kd,n] = CB[kd*K + n]
// Software-pipelined like the conv loop; 32-bit indices.
// ---------------------------------------------------------------------------
__global__ __launch_bounds__(128)
void scores_wmma(const float* __restrict__ Ze, const float* __restrict__ CB,
                 const float* __restrict__ cbn, float* __restrict__ S,
                 int D, int T, int K)
{
    const int lane    = threadIdx.x & 31;
    const int wave    = threadIdx.x >> 5;
    const int laneLow = lane & 15;
    const int laneHi  = lane >> 4;
    const int mBase   = (blockIdx.y * 4 + wave) * 16;   // token index t
    const int nBase   = blockIdx.x * 64;                // code index k

    v8f acc[4] = {};

    unsigned zoff = (unsigned)(laneHi * 2) * (unsigned)T + (unsigned)(mBase + laneLow);
    unsigned coff = (unsigned)(laneHi * 2) * (unsigned)K + (unsigned)(nBase + laneLow);
    const unsigned zstep = 4u * (unsigned)T;
    const unsigned cstep = 4u * (unsigned)K;

    v2f a;
    float bx[4], by[4];
    a.x = Ze[zoff];
    a.y = Ze[zoff + (unsigned)T];
#pragma unroll
    for (int nt = 0; nt < 4; ++nt) {
        bx[nt] = CB[coff + (unsigned)(nt * 16)];
        by[nt] = CB[coff + (unsigned)(nt * 16 + K)];
    }

    for (int kk = 4; kk < D; kk += 4) {
        zoff += zstep;
        coff += cstep;
        v2f an;
        float bxn[4], byn[4];
        an.x = Ze[zoff];
        an.y = Ze[zoff + (unsigned)T];
#pragma unroll
        for (int nt = 0; nt < 4; ++nt) {
            bxn[nt] = CB[coff + (unsigned)(nt * 16)];
            byn[nt] = CB[coff + (unsigned)(nt * 16 + K)];
        }
#pragma unroll
        for (int nt = 0; nt < 4; ++nt) {
            v2f b; b.x = bx[nt]; b.y = by[nt];
            acc[nt] = __builtin_amdgcn_wmma_f32_16x16x4_f32(
                false, a, false, b, (short)0, acc[nt], false, false);
        }
        a = an;
#pragma unroll
        for (int nt = 0; nt < 4; ++nt) { bx[nt] = bxn[nt]; by[nt] = byn[nt]; }
    }
#pragma unroll
    for (int nt = 0; nt < 4; ++nt) {
        v2f b; b.x = bx[nt]; b.y = by[nt];
        acc[nt] = __builtin_amdgcn_wmma_f32_16x16x4_f32(
            false, a, false, b, (short)0, acc[nt], false, false);
    }

#pragma unroll
    for (int nt = 0; nt < 4; ++nt) {
        const int col = nBase + nt * 16 + laneLow;
        const float cn = cbn[col];
#pragma unroll
        for (int r = 0; r < 8; ++r) {
            const int row = mBase + laneHi * 8 + r;
            S[(size_t)row * (size_t)K + col] = cn - 2.0f * acc[nt][r];
        }
    }
}

// --------------------------- small utility kernels -------------------------

__global__ void cbnorm_kernel(const float* __restrict__ CB, float* __restrict__ cbn,
                              int D, int K)
{
    const int k = blockIdx.x * blockDim.x + threadIdx.x;
    if (k >= K) return;
    float s = 0.0f;
    for (int d = 0; d < D; ++d) {
        const float v = CB[(size_t)d * (size_t)K + k];
        s += v * v;
    }
    cbn[k] = s;
}

__global__ void argmin_rows(const float* __restrict__ S, int T, int K,
                            int* __restrict__ idxI, float* __restrict__ idxF)
{
    const int t = blockIdx.x * blockDim.x + threadIdx.x;
    if (t >= T) return;
    const float* __restrict__ row = S + (size_t)t * (size_t)K;
    float best = row[0];
    int   bi   = 0;
    for (int k = 1; k < K; ++k) {
        const float v = row[k];
        if (v < best) { best = v; bi = k; }   // strict < keeps first index on ties
    }
    idxI[t] = bi;
    idxF[t] = (float)bi;
}

// z_q[d,t] = CB[d, idx[t]]  (straight-through value == gathered code)
__global__ void gather_zq(const float* __restrict__ CB, const int* __restrict__ idx,
                          float* __restrict__ Zq, int D, int T, int K)
{
    const int i = blockIdx.x * blockDim.x + threadIdx.x;
    if (i >= D * T) return;
    const int d = i / T;
    const int t = i - d * T;
    Zq[(size_t)d * (size_t)T + t] = CB[(size_t)d * (size_t)K + idx[t]];
}

__global__ void counts_kernel(const int* __restrict__ idx, float* __restrict__ counts,
                              int T)
{
    const int t = blockIdx.x * blockDim.x + threadIdx.x;
    if (t >= T) return;
    atomicAdd(&counts[idx[t]], 1.0f);
}

// scat[d, idx[t]] += Ze[d, t]   (flat.T @ onehot)
__global__ void scatter_avg(const float* __restrict__ Ze, const int* __restrict__ idx,
                            float* __restrict__ scat, int D, int T, int K)
{
    const int i = blockIdx.x * blockDim.x + threadIdx.x;
    if (i >= D * T) return;
    const int d = i / T;
    const int t = i - d * T;
    atomicAdd(&scat[(size_t)d * (size_t)K + idx[t]], Ze[(size_t)d * (size_t)T + t]);
}

__global__ void new_cs_raw(const float* __restrict__ clsz, const float* __restrict__ counts,
                           float* __restrict__ out_cs, int K)
{
    const int k = blockIdx.x * blockDim.x + threadIdx.x;
    if (k >= K) return;
    out_cs[k] = DECAY * clsz[k] + (1.0f - DECAY) * counts[k];
}

// single block, K==blockDim.x: n = sum(cs); cs = (cs+eps)/(n+K*eps)*n
__global__ void normalize_cs(float* __restrict__ cs, int K)
{
    __shared__ float red[1024];
    const int k = threadIdx.x;
    const float v = cs[k];
    red[k] = v;
    __syncthreads();
    for (int s = 512; s > 0; s >>= 1) {
        if (k < s) red[k] += red[k + s];
        __syncthreads();
    }
    const float n = red[0];
    cs[k] = (v + EPSV) / (n + (float)K * EPSV) * n;
}

__global__ void new_avg_kernel(const float* __restrict__ cb_avg, const float* __restrict__ scat,
                               float* __restrict__ out_avg, int N)
{
    const int i = blockIdx.x * blockDim.x + threadIdx.x;
    if (i >= N) return;
    out_avg[i] = DECAY * cb_avg[i] + (1.0f - DECAY) * scat[i];
}

// new_cb = OLD codebook_avg / normalized new_cs   (matches reference semantics)
__global__ void new_cb_kernel(const float* __restrict__ cb_avg, const float* __restrict__ cs,
                              float* __restrict__ out_cb, int D, int K)
{
    const int i = blockIdx.x * blockDim.x + threadIdx.x;
    if (i >= D * K) return;
    const int k = i % K;
    out_cb[i] = cb_avg[i] / cs[k];
}

// ------------------------------- host side ---------------------------------

static inline void launch_conv(const float* X, const float* W, const float* B,
                               const float* Res, float* Y,
                               int Cin, int Cout, int Lin, int Lout,
                               int ks, int stride, int relu, int rep2,
                               hipStream_t s)
{
    dim3 g((Lout + 63) / 64, (Cout + 63) / 64);
    if (ks == 3 && stride == 2)
        conv1d_wmma<3, 2, 0><<<g, 128, 0, s>>>(X, W, B, Res, Y, Cin, Cout, Lin, Lout, relu);
    else if (ks == 3 && rep2)
        conv1d_wmma<3, 1, 1><<<g, 128, 0, s>>>(X, W, B, Res, Y, Cin, Cout, Lin, Lout, relu);
    else if (ks == 3)
        conv1d_wmma<3, 1, 0><<<g, 128, 0, s>>>(X, W, B, Res, Y, Cin, Cout, Lin, Lout, relu);
    else
        conv1d_wmma<1, 1, 0><<<g, 128, 0, s>>>(X, W, B, Res, Y, Cin, Cout, Lin, Lout, relu);
}

// resblock on buf (HIDxL) in place; t1,t2 are HIDxL temps
static inline void launch_resblock(float* buf, float* t1, float* t2,
                                   const float* w1, const float* b1,
                                   const float* w2, const float* b2,
                                   const float* w3, const float* b3,
                                   int L, hipStream_t s)
{
    launch_conv(buf, w1, b1, nullptr, t1, 1024, 1024, L, L, 3, 1, 1, 0, s);
    launch_conv(t1,  w2, b2, nullptr, t2, 1024, 1024, L, L, 3, 1, 1, 0, s);
    launch_conv(t2,  w3, b3, buf,     buf, 1024, 1024, L, L, 1, 1, 0, 0, s);
}

extern "C" void kernel_launch(void* const* d_in, const int* in_sizes, int n_in,
                              void* d_out, int out_size, void* d_ws, size_t ws_size,
                              hipStream_t stream)
{
    (void)in_sizes; (void)n_in; (void)out_size; (void)ws_size;
    // Input order: setup_inputs() dict, nested dicts in insertion order.
    #define IN(i) ((const float*)d_in[(i)])
    const float* x = IN(0);
    // encoder
    const float *e_c1w = IN(1),  *e_c1b = IN(2);
    const float *e_c2w = IN(3),  *e_c2b = IN(4);
    const int ER[3] = {5, 11, 17};                     // enc resblock bases
    const float *e_c3w = IN(23), *e_c3b = IN(24);
    // decoder
    const float *d_c1w = IN(25), *d_c1b = IN(26);
    const int DR[3] = {27, 33, 39};                    // dec resblock bases
    const float *d_c2w = IN(45), *d_c2b = IN(46);
    const float *d_c3w = IN(47), *d_c3b = IN(48);
    const float *d_c4w = IN(49), *d_c4b = IN(50);
    const float *codebook     = IN(51);
    const float *codebook_avg = IN(52);
    const float *cluster_size = IN(53);

    const int T  = 32768;     // mel frames
    const int Tq = 8192;      // latent frames (T/4)
    const int D  = 512;       // codebook dim
    const int K  = 1024;      // codes

    // d_out layout (floats), tuple order (z_e, z_q, new_cs, new_avg, new_cb, idx, y)
    float* out = (float*)d_out;
    float* o_ze  = out;                                  // 512*8192
    float* o_zq  = o_ze  + (size_t)D * Tq;               // 512*8192
    float* o_cs  = o_zq  + (size_t)D * Tq;               // 1024
    float* o_avg = o_cs  + K;                            // 512*1024
    float* o_cb  = o_avg + (size_t)D * K;                // 512*1024
    float* o_idx = o_cb  + (size_t)D * K;                // 8192
    float* o_y   = o_idx + Tq;                           // 80*32768

    // workspace layout (floats)
    float* ws   = (float*)d_ws;
    float* bufA = ws;                                    // 16M floats (64MB)
    float* bufB = bufA + (size_t)16 * 1024 * 1024;       // 16M floats
    float* bufT = bufB + (size_t)16 * 1024 * 1024;       // 8M floats (scores/resblock temp)
    float* cbn  = bufT + (size_t)8  * 1024 * 1024;       // 1024
    float* cnts = cbn  + K;                              // 1024
    float* scat = cnts + K;                              // 512*1024
    int*   idxI = (int*)(scat + (size_t)D * K);          // 8192

    // ---------------- encoder ----------------
    launch_conv(x,    e_c1w, e_c1b, nullptr, bufA,  80,  512, T,     T / 2, 3, 2, 1, 0, stream);
    launch_conv(bufA, e_c2w, e_c2b, nullptr, bufB, 512, 1024, T / 2, Tq,    3, 2, 1, 0, stream);
    for (int r = 0; r < 3; ++r)
        launch_resblock(bufB, bufT, bufA,
                        IN(ER[r] + 0), IN(ER[r] + 1), IN(ER[r] + 2),
                        IN(ER[r] + 3), IN(ER[r] + 4), IN(ER[r] + 5), Tq, stream);
    launch_conv(bufB, e_c3w, e_c3b, nullptr, o_ze, 1024, 512, Tq, Tq, 1, 1, 0, 0, stream);

    // ---------------- quantizer ----------------
    cbnorm_kernel<<<dim3((K + 255) / 256), 256, 0, stream>>>(codebook, cbn, D, K);
    {
        dim3 g(K / 64, Tq / 64);
        scores_wmma<<<g, 128, 0, stream>>>(o_ze, codebook, cbn, bufT, D, Tq, K);
    }
    argmin_rows<<<dim3((Tq + 255) / 256), 256, 0, stream>>>(bufT, Tq, K, idxI, o_idx);
    gather_zq<<<dim3((D * Tq + 255) / 256), 256, 0, stream>>>(codebook, idxI, o_zq, D, Tq, K);

    hipMemsetAsync(cnts, 0, (size_t)(K + D * K) * sizeof(float), stream);  // counts + scat
    counts_kernel<<<dim3((Tq + 255) / 256), 256, 0, stream>>>(idxI, cnts, Tq);
    scatter_avg<<<dim3((D * Tq + 255) / 256), 256, 0, stream>>>(o_ze, idxI, scat, D, Tq, K);

    new_cs_raw<<<dim3((K + 255) / 256), 256, 0, stream>>>(cluster_size, cnts, o_cs, K);
    normalize_cs<<<1, 1024, 0, stream>>>(o_cs, K);
    new_avg_kernel<<<dim3((D * K + 255) / 256), 256, 0, stream>>>(codebook_avg, scat, o_avg, D * K);
    new_cb_kernel<<<dim3((D * K + 255) / 256), 256, 0, stream>>>(codebook_avg, o_cs, o_cb, D, K);

    // ---------------- decoder ----------------
    launch_conv(o_zq, d_c1w, d_c1b, nullptr, bufA, 512, 1024, Tq, Tq, 1, 1, 0, 0, stream);
    for (int r = 0; r < 3; ++r)
        launch_resblock(bufA, bufT, bufB,
                        IN(DR[r] + 0), IN(DR[r] + 1), IN(DR[r] + 2),
                        IN(DR[r] + 3), IN(DR[r] + 4), IN(DR[r] + 5), Tq, stream);
    // repeat2 folds the 2x nearest upsample into the conv's input indexing
    launch_conv(bufA, d_c2w, d_c2b, nullptr, bufB, 1024, 1024, Tq,     T / 2, 3, 1, 1, 1, stream);
    launch_conv(bufB, d_c3w, d_c3b, nullptr, bufA, 1024,  512, T / 2,  T,     3, 1, 1, 1, stream);
    launch_conv(bufA, d_c4w, d_c4b, nullptr, o_y,   512,   80, T,      T,     1, 1, 0, 0, stream);
    #undef IN
}